// GINwithClassifier_9380208574710
// MI455X (gfx1250) — compile-verified
//
#include <hip/hip_runtime.h>
#include <hip/hip_bf16.h>

// ---------------------------------------------------------------------------
// GIN (5 layers) + global-add-pool + MLP classifier with BatchNorm, fp32.
// Dense GEMMs: V_WMMA_F32_16X16X4_F32, 64x32 output per wave, B stripes
// staged in LDS by the Tensor Data Mover (tensor_load_to_lds).
// ---------------------------------------------------------------------------

#define NNODES  50000
#define NEDGES  800000
#define NGRAPHS 512
#define FIN     128
#define HDIM    300
#define HP      304      // HDIM padded to multiple of 16
#define CDIM    10
#define CP      16       // CDIM padded
#define MWAVE   4        // m-tiles (of 16 rows) per wave
#define NWAVE   2        // n-tiles (of 16 cols) per wave

typedef __attribute__((ext_vector_type(2))) float v2f;
typedef __attribute__((ext_vector_type(8))) float v8f;
typedef __attribute__((ext_vector_type(4))) unsigned int v4u;
typedef __attribute__((ext_vector_type(8))) int v8i;
typedef __attribute__((ext_vector_type(4))) int v4i;

// ---------------------------------------------------------------------------
// Generic zero-padding repack: dst[prows x pcols] <- src[rows x cols], 0 pad.
// ---------------------------------------------------------------------------
__global__ void gin_pack_pad(const float* __restrict__ src, float* __restrict__ dst,
                             int rows, int cols, int prows, int pcols) {
  int idx = blockIdx.x * blockDim.x + threadIdx.x;
  int total = prows * pcols;
  if (idx >= total) return;
  int r = idx / pcols, c = idx % pcols;
  dst[idx] = (r < rows && c < cols) ? src[r * cols + c] : 0.0f;
}

__global__ void gin_copy(const float* __restrict__ a, float* __restrict__ b, long n) {
  long i = (long)blockIdx.x * blockDim.x + threadIdx.x;
  if (i < n) b[i] = a[i];
}

__global__ void gin_zero(float* __restrict__ a, long n) {
  long i = (long)blockIdx.x * blockDim.x + threadIdx.x;
  if (i < n) a[i] = 0.0f;
}

// ---------------------------------------------------------------------------
// Edge scatter-add: z[dst[e], f] += h[src[e], f]   (HW fp32 atomics, hits L2)
// ---------------------------------------------------------------------------
__global__ void gin_edge_scatter(const float* __restrict__ h, float* __restrict__ z,
                                 const int* __restrict__ src, const int* __restrict__ dst,
                                 int feat, int ld) {
  int e = blockIdx.x;
  int f = blockIdx.y * blockDim.x + threadIdx.x;
  if (f >= feat) return;
  int u = src[e], v = dst[e];
  unsafeAtomicAdd(&z[(size_t)v * ld + f], h[(size_t)u * ld + f]);
}

// Pool: g[batch[v], f] += h[v, f]
__global__ void gin_pool_scatter(const float* __restrict__ h, float* __restrict__ g,
                                 const int* __restrict__ batch, int feat, int ld) {
  int v = blockIdx.x;
  int f = blockIdx.y * blockDim.x + threadIdx.x;
  if (f >= feat) return;
  unsafeAtomicAdd(&g[(size_t)batch[v] * ld + f], h[(size_t)v * ld + f]);
}

// ---------------------------------------------------------------------------
// WMMA fp32 GEMM:  C = [relu]( A[MxK] @ B[KxN] + bias[N] )
// Block = 256 threads = 8 waves.  Wave -> 4 m-tiles x 2 n-tiles (64x32 out).
// grid = (ceil(M/16 / 32), ceil(N/16 / 2)), dynamic LDS = K*32*4 bytes.
//
// B stripe [K x 32] is DMA'd into LDS once per block by the Tensor Data
// Mover; tensor_dim0 = min(N-n0, 32) makes TDM zero-fill columns past the
// matrix edge (OOB reads return zero), handling the ragged last pair.
// Per K-block of 4: 4 A b64 loads + 2 ds_2addr loads -> 8 WMMAs.
//
// V_WMMA_F32_16X16X4_F32 operand layouts (ISA 7.12.2):
//   A (16x4): lanes 0-15 row M=lane,   VGPR0=K0 VGPR1=K1;
//             lanes 16-31 row M=lane-16, VGPR0=K2 VGPR1=K3.
//   B (4x16): lanes 0-15 col N=lane,   VGPR0=K0 VGPR1=K1;
//             lanes 16-31 col N=lane-16, VGPR0=K2 VGPR1=K3.
//   D (16x16): VGPR v: lanes 0-15 row v, lanes 16-31 row v+8.
// ---------------------------------------------------------------------------
__global__ void gin_wmma_gemm(const float* __restrict__ A, const float* __restrict__ B,
                              const float* __restrict__ bias, float* __restrict__ C,
                              int M, int N, int K, int lda, int ldb, int ldc,
                              int nvalid, int do_relu) {
  extern __shared__ float Bs[];             // [K x 32], k-major
  const int lane = threadIdx.x & 31;
  const int wave = threadIdx.x >> 5;        // 0..7
  const int n0 = blockIdx.y * (16 * NWAVE);
  const int half = lane >> 4;               // 0 or 1
  const int l15  = lane & 15;

  // ---- TDM: DMA the B stripe [K x 32] into LDS (wave 0 issues) ----
  if (wave == 0) {
    unsigned long long ga = (unsigned long long)(uintptr_t)(B + n0);
    unsigned ldsOff = (unsigned)(uintptr_t)(void*)Bs;   // low 32 bits = LDS offset
    int avail = N - n0; if (avail > 32) avail = 32;     // valid cols; rest zero-fill
    v4u g0;
    g0.x = 1u;                                          // count=1 (valid user D#)
    g0.y = ldsOff;                                      // lds_addr
    g0.z = (unsigned)(ga & 0xffffffffu);                // global_addr[31:0]
    g0.w = (unsigned)((ga >> 32) & 0x01ffffffu) | 0x80000000u; // addr[56:32] | type=2
    v8i g1;
    g1[0] = 0x00020000;                       // data_size=2 -> 4-byte elements
    g1[1] = (int)((unsigned)avail << 16);     // tensor_dim0[15:0] in bits[63:48]
    g1[2] = (int)((unsigned)K << 16);         // tensor_dim1[15:0]=K in bits[95:80]
    g1[3] = (int)(32u << 16);                 // tile_dim0=32 in bits[127:112]
    g1[4] = K;                                // tile_dim1=K; tile_dim2=0
    g1[5] = ldb;                              // tensor_dim0_stride[31:0] (elements)
    g1[6] = 0;
    g1[7] = 0;
    v4i g2 = {0, 0, 0, 0};
    v4i g3 = {0, 0, 0, 0};
    v8i g4 = {0, 0, 0, 0, 0, 0, 0, 0};        // extra group (clang-23 6-arg form)
    __builtin_amdgcn_tensor_load_to_lds(g0, g1, g2, g3, g4, 0);
    __builtin_amdgcn_s_wait_tensorcnt(0);
  }
  __syncthreads();

  // ---- per-wave m-tiles ----
  const int mtiles = M >> 4;
  const int mt0 = (blockIdx.x * 8 + wave) * MWAVE;      // wave-uniform
  const float* arow[MWAVE];
  #pragma unroll
  for (int j = 0; j < MWAVE; ++j) {
    const int mt = mt0 + j;
    const int r = (mt < mtiles) ? (mt * 16 + l15) : l15;  // clamp keeps loads legal
    arow[j] = A + (size_t)r * lda;
  }

  v8f acc[NWAVE][MWAVE] = {};
  #pragma unroll 2
  for (int kb = 0; kb < K; kb += 4) {
    const int ka = kb + 2 * half;
    v2f b0, b1;
    b0.x = Bs[ka * 32 + l15];
    b0.y = Bs[(ka + 1) * 32 + l15];
    b1.x = Bs[ka * 32 + 16 + l15];
    b1.y = Bs[(ka + 1) * 32 + 16 + l15];
    #pragma unroll
    for (int j = 0; j < MWAVE; ++j) {
      v2f a;
      a.x = arow[j][ka];
      a.y = arow[j][ka + 1];
      acc[0][j] = __builtin_amdgcn_wmma_f32_16x16x4_f32(
          false, a, false, b0, (short)0, acc[0][j], false, false);
      acc[1][j] = __builtin_amdgcn_wmma_f32_16x16x4_f32(
          false, a, false, b1, (short)0, acc[1][j], false, false);
    }
  }

  // ---- epilogue: bias (+ReLU), store valid tiles/columns ----
  #pragma unroll
  for (int t = 0; t < NWAVE; ++t) {
    const int col = n0 + t * 16 + l15;
    const float bv = (col < N) ? bias[col] : 0.0f;
    #pragma unroll
    for (int j = 0; j < MWAVE; ++j) {
      const int mt = mt0 + j;
      if (mt < mtiles && col < nvalid) {
        float* crow = C + (size_t)(mt * 16 + 8 * half) * ldc + col;
        #pragma unroll
        for (int v = 0; v < 8; ++v) {
          float x = acc[t][j][v] + bv;
          if (do_relu) x = fmaxf(x, 0.0f);
          crow[(size_t)v * ldc] = x;
        }
      }
    }
  }
}

// ---------------------------------------------------------------------------
// BatchNorm (biased var, torch semantics) + ReLU over [NGRAPHS x HP] columns.
// ---------------------------------------------------------------------------
__global__ void gin_bn_relu(const float* __restrict__ z, float* __restrict__ out,
                            const float* __restrict__ gamma, const float* __restrict__ beta) {
  const int c = blockIdx.x;     // 0..HP-1
  const int tid = threadIdx.x;  // 256
  __shared__ float ssum[256];
  __shared__ float ssq[256];
  float sum = 0.0f, sq = 0.0f;
  if (c < HDIM) {
    for (int r = tid; r < NGRAPHS; r += 256) {
      float v = z[(size_t)r * HP + c];
      sum += v; sq += v * v;
    }
  }
  ssum[tid] = sum; ssq[tid] = sq;
  __syncthreads();
  for (int s = 128; s > 0; s >>= 1) {
    if (tid < s) { ssum[tid] += ssum[tid + s]; ssq[tid] += ssq[tid + s]; }
    __syncthreads();
  }
  const float inv_n = 1.0f / (float)NGRAPHS;
  const float mean = ssum[0] * inv_n;
  const float var  = ssq[0] * inv_n - mean * mean;
  float scale = 0.0f, shift = 0.0f;
  if (c < HDIM) {
    scale = rsqrtf(var + 1e-5f) * gamma[c];
    shift = beta[c] - mean * scale;
  }
  for (int r = tid; r < NGRAPHS; r += 256) {
    float v = 0.0f;
    if (c < HDIM) v = fmaxf(z[(size_t)r * HP + c] * scale + shift, 0.0f);
    out[(size_t)r * HP + c] = v;
  }
}

// ---------------------------------------------------------------------------
// Host-side launch
// ---------------------------------------------------------------------------
static inline int cdiv(long a, long b) { return (int)((a + b - 1) / b); }

static inline void launch_gemm(const float* A, const float* B, const float* bias, float* C,
                               int M, int N, int K, int lda, int ldb, int ldc,
                               int nvalid, int relu, hipStream_t s) {
  dim3 grid(cdiv(M / 16, 8 * MWAVE), cdiv(N / 16, NWAVE));
  size_t ldsB = (size_t)K * 32 * sizeof(float);
  gin_wmma_gemm<<<grid, 256, ldsB, s>>>(A, B, bias, C, M, N, K, lda, ldb, ldc, nvalid, relu);
}

static inline void launch_pack(const float* src, float* dst, int r, int c, int pr, int pc,
                               hipStream_t s) {
  int total = pr * pc;
  gin_pack_pad<<<cdiv(total, 256), 256, 0, s>>>(src, dst, r, c, pr, pc);
}

extern "C" void kernel_launch(void* const* d_in, const int* in_sizes, int n_in,
                              void* d_out, int out_size, void* d_ws, size_t ws_size,
                              hipStream_t stream) {
  // ---- inputs (setup_inputs dict order) ----
  const float* x        = (const float*)d_in[0];   // [NNODES, FIN]
  const int*   edge     = (const int*)d_in[1];     // [2, NEDGES]
  const int*   batch    = (const int*)d_in[2];     // [NNODES]
  const float* w1_0     = (const float*)d_in[3];   // [FIN, HDIM]
  const float* b1_0     = (const float*)d_in[4];
  const float* w2_0     = (const float*)d_in[5];   // [HDIM, HDIM]
  const float* b2_0     = (const float*)d_in[6];
  const float* w1_rest  = (const float*)d_in[7];   // [4, HDIM, HDIM]
  const float* b1_rest  = (const float*)d_in[8];   // [4, HDIM]
  const float* w2_rest  = (const float*)d_in[9];
  const float* b2_rest  = (const float*)d_in[10];
  const float* wc1      = (const float*)d_in[11];  // [HDIM, HDIM]
  const float* bc1      = (const float*)d_in[12];
  const float* bn_g     = (const float*)d_in[13];
  const float* bn_b     = (const float*)d_in[14];
  const float* wc2      = (const float*)d_in[15];  // [HDIM, CDIM]
  const float* bc2      = (const float*)d_in[16];
  float* out            = (float*)d_out;           // [NGRAPHS, CDIM]

  const int* e_src = edge;            // edge_index[0]
  const int* e_dst = edge + NEDGES;   // edge_index[1]

  // ---- workspace carve-up ----
  char* p = (char*)d_ws;
  auto alloc = [&](size_t nfloats) {
    float* r = (float*)p;
    p += ((nfloats * sizeof(float) + 255) / 256) * 256;
    return r;
  };
  const size_t SZ_NODE = (size_t)NNODES * HP;
  float* Hb    = alloc(SZ_NODE);          // current node features (padded)
  float* Zb    = alloc(SZ_NODE);          // h + aggregated messages
  float* Tb    = alloc(SZ_NODE);          // inner-MLP hidden
  float* w1_0p = alloc((size_t)FIN * HP);
  float* w2_0p = alloc((size_t)HP * HP);
  float* w1p   = alloc((size_t)4 * HP * HP);
  float* w2p   = alloc((size_t)4 * HP * HP);
  float* wc1p  = alloc((size_t)HP * HP);
  float* wc2p  = alloc((size_t)HP * CP);
  float* b1_0p = alloc(HP);
  float* b2_0p = alloc(HP);
  float* b1p   = alloc(4 * HP);
  float* b2p   = alloc(4 * HP);
  float* bc1p  = alloc(HP);
  float* bc2p  = alloc(CP);
  float* Gb    = alloc((size_t)NGRAPHS * HP);
  float* Z1b   = alloc((size_t)NGRAPHS * HP);
  float* Z2b   = alloc((size_t)NGRAPHS * HP);

  // ---- repack weights/biases with zero padding (deterministic per call) ----
  launch_pack(w1_0, w1_0p, FIN,  HDIM, FIN, HP, stream);
  launch_pack(w2_0, w2_0p, HDIM, HDIM, HP,  HP, stream);
  for (int i = 0; i < 4; ++i) {
    launch_pack(w1_rest + (size_t)i * HDIM * HDIM, w1p + (size_t)i * HP * HP,
                HDIM, HDIM, HP, HP, stream);
    launch_pack(w2_rest + (size_t)i * HDIM * HDIM, w2p + (size_t)i * HP * HP,
                HDIM, HDIM, HP, HP, stream);
    launch_pack(b1_rest + (size_t)i * HDIM, b1p + (size_t)i * HP, 1, HDIM, 1, HP, stream);
    launch_pack(b2_rest + (size_t)i * HDIM, b2p + (size_t)i * HP, 1, HDIM, 1, HP, stream);
  }
  launch_pack(wc1, wc1p, HDIM, HDIM, HP, HP, stream);
  launch_pack(wc2, wc2p, HDIM, CDIM, HP, CP, stream);
  launch_pack(b1_0, b1_0p, 1, HDIM, 1, HP, stream);
  launch_pack(b2_0, b2_0p, 1, HDIM, 1, HP, stream);
  launch_pack(bc1,  bc1p,  1, HDIM, 1, HP, stream);
  launch_pack(bc2,  bc2p,  1, CDIM, 1, CP, stream);

  // ---- GIN layer 0: x (ld=FIN) ----
  {
    long n = (long)NNODES * FIN;
    gin_copy<<<cdiv(n, 256), 256, 0, stream>>>(x, Zb, n);   // Z = x (ld FIN)
    dim3 sg(NEDGES, cdiv(FIN, 128));
    gin_edge_scatter<<<sg, 128, 0, stream>>>(x, Zb, e_src, e_dst, FIN, FIN);
    launch_gemm(Zb, w1_0p, b1_0p, Tb, NNODES, HP, FIN, FIN, HP, HP, HP, 1, stream);
    launch_gemm(Tb, w2_0p, b2_0p, Hb, NNODES, HP, HP,  HP,  HP, HP, HP, 1, stream);
  }

  // ---- GIN layers 1..4 (ld=HP) ----
  for (int i = 0; i < 4; ++i) {
    long n = (long)NNODES * HP;
    gin_copy<<<cdiv(n, 256), 256, 0, stream>>>(Hb, Zb, n);
    dim3 sg(NEDGES, cdiv(HP, 128));
    gin_edge_scatter<<<sg, 128, 0, stream>>>(Hb, Zb, e_src, e_dst, HP, HP);
    launch_gemm(Zb, w1p + (size_t)i * HP * HP, b1p + (size_t)i * HP, Tb,
                NNODES, HP, HP, HP, HP, HP, HP, 1, stream);
    launch_gemm(Tb, w2p + (size_t)i * HP * HP, b2p + (size_t)i * HP, Hb,
                NNODES, HP, HP, HP, HP, HP, HP, 1, stream);
  }

  // ---- global add pool ----
  {
    long n = (long)NGRAPHS * HP;
    gin_zero<<<cdiv(n, 256), 256, 0, stream>>>(Gb, n);
    dim3 pg(NNODES, cdiv(HP, 128));
    gin_pool_scatter<<<pg, 128, 0, stream>>>(Hb, Gb, batch, HP, HP);
  }

  // ---- classifier: Linear -> BN -> ReLU -> Linear ----
  launch_gemm(Gb, wc1p, bc1p, Z1b, NGRAPHS, HP, HP, HP, HP, HP, HP, 0, stream);
  gin_bn_relu<<<HP, 256, 0, stream>>>(Z1b, Z2b, bn_g, bn_b);
  launch_gemm(Z2b, wc2p, bc2p, out, NGRAPHS, CP, HP, HP, CP, CDIM, CDIM, 0, stream);
}